// VectorQuantizer_25709674233923
// MI455X (gfx1250) — compile-verified
//
#include <hip/hip_runtime.h>
#include <hip/hip_bf16.h>
#include <math.h>

// ---------------------------------------------------------------------------
// VQ-VAE vector quantizer for MI455X (gfx1250, wave32)
//   z : (64, 64, 32, 32) f32  (B, C, H, W), C = 64
//   w : (1024, 64) f32 codebook
// Outputs (flat in d_out): [loss(1), z_q(64*64*32*32 BCHW), perplexity(1)]
// Distance matmul uses V_WMMA_F32_16X16X4_F32 (fp32 precision preserved --
// codebook entries are ~1e-3 and argmin gaps would drown in bf16 rounding).
// ---------------------------------------------------------------------------

#define N_E      1024
#define C_DIM    64
#define N_VEC    65536        // 64 * 32 * 32
#define HW       1024         // 32 * 32
#define TOT_ELEM 4194304      // N_VEC * C_DIM

typedef __attribute__((ext_vector_type(2))) float v2f;
typedef __attribute__((ext_vector_type(8))) float v8f;

// LDS-only split barrier: retire own DS ops, signal, wait. Unlike
// __syncthreads() this does NOT flush loadcnt, so register prefetches
// (global_load_b128) legitimately stay in flight across the barrier.
#define LDS_BARRIER() \
  asm volatile("s_wait_dscnt 0x0\n\t" \
               "s_barrier_signal -1\n\t" \
               "s_barrier_wait -1" ::: "memory")

// ---------------------------------------------------------------------------
// Kernel 1: transpose codebook (for coalesced B loads), codeword norms,
//           zero histogram + loss accumulator. 1024 threads.
// ---------------------------------------------------------------------------
__global__ void vq_prep(const float* __restrict__ w,
                        float* __restrict__ wt,      // (64, 1024) transposed
                        float* __restrict__ wn,      // (1024,) ||w_j||^2
                        int*   __restrict__ hist,    // (1024,)
                        float* __restrict__ lossAcc) {
  int j = blockIdx.x * blockDim.x + threadIdx.x;
  if (j >= N_E) return;
  float s = 0.0f;
#pragma unroll
  for (int c = 0; c < C_DIM; ++c) {
    float v = w[(size_t)j * C_DIM + c];
    s += v * v;
    wt[(size_t)c * N_E + j] = v;   // coalesced across j
  }
  wn[j] = s;
  hist[j] = 0;
  if (j == 0) *lossAcc = 0.0f;
}

// ---------------------------------------------------------------------------
// Kernel 2: WMMA distance matmul + per-row argmin.
// Block = 256 threads (8 waves). Wave owns TWO 16-row M tiles (32 rows) ->
// two accumulator chains sharing each B fragment; block = 256 rows.
// Grid = 256 blocks. d'(m,j) = ||w_j||^2 - 2 * <z_m, w_j>  (||z||^2 dropped).
// ---------------------------------------------------------------------------
__global__ void __launch_bounds__(256)
vq_argmin(const float* __restrict__ z,
          const float* __restrict__ wt,
          const float* __restrict__ wn,
          int* __restrict__ idxOut,
          int* __restrict__ hist) {
  __shared__ float sB[16 * 68];            // 16 codewords x 64 dims, pad 68

  const int tid    = threadIdx.x;
  const int wave   = tid >> 5;             // 0..7
  const int lane   = tid & 31;
  const int laneLo = lane & 15;            // column / M within half
  const int laneHi = lane >> 4;            // K half select

  // staging decomposition: thread -> (c row, 4 consecutive jl) float4 load
  const int stC  = tid >> 2;               // 0..63
  const int stJl = (tid & 3) * 4;          // 0,4,8,12

  // --- Preload A fragments: 2 tiles x 16 rows x K=64 fp32 (64 VGPRs) ---
  const int nBase = blockIdx.x * 256 + wave * 32;
  const int n0    = nBase + laneLo;              // tile 0 row (M = laneLo)
  const int n1    = nBase + 16 + laneLo;         // tile 1 row
  const int bimg  = n0 >> 10;                    // same batch for both tiles
  const float* zrow0 = z + ((size_t)bimg * C_DIM) * HW + (n0 & (HW - 1));
  const float* zrow1 = z + ((size_t)bimg * C_DIM) * HW + (n1 & (HW - 1));

  v2f afragA[16], afragB[16];
#pragma unroll
  for (int kk = 0; kk < 16; ++kk) {
    const int c0 = 4 * kk + 2 * laneHi;    // A layout: VGPR0 K={0,2}, VGPR1 K={1,3}
    afragA[kk][0] = zrow0[(size_t)c0 * HW];
    afragA[kk][1] = zrow0[(size_t)(c0 + 1) * HW];
    afragB[kk][0] = zrow1[(size_t)c0 * HW];
    afragB[kk][1] = zrow1[(size_t)(c0 + 1) * HW];
  }

  float minv0[8], minv1[8];
  int   mini0[8], mini1[8];
#pragma unroll
  for (int r = 0; r < 8; ++r) {
    minv0[r] = 3.4e38f; mini0[r] = 0;
    minv1[r] = 3.4e38f; mini1[r] = 0;
  }

  // prefetch first staging tile into registers
  float4 stv = *(const float4*)&wt[(size_t)stC * N_E + stJl];

  // --- Sweep codebook in tiles of 16 codewords ---
  for (int j0 = 0; j0 < N_E; j0 += 16) {
    LDS_BARRIER();                         // readers of previous tile done
    // scatter staged tile: bank = ((jl+k)*68 + c) % 64 conflict-free
    sB[(stJl + 0) * 68 + stC] = stv.x;
    sB[(stJl + 1) * 68 + stC] = stv.y;
    sB[(stJl + 2) * 68 + stC] = stv.z;
    sB[(stJl + 3) * 68 + stC] = stv.w;
    // prefetch NEXT tile; stays in flight across the split barrier and is
    // waited only at next iteration's ds_store -> hidden behind 32 WMMAs
    const int jn = (j0 + 16) & (N_E - 1);        // wraps to 0 on last iter
    stv = *(const float4*)&wt[(size_t)stC * N_E + jn + stJl];
    LDS_BARRIER();                         // staged tile visible block-wide

    v8f acc0 = {}, acc1 = {};
#pragma unroll
    for (int kk = 0; kk < 16; ++kk) {
      const int c0 = 4 * kk + 2 * laneHi;  // B layout mirrors A K-half split
      const float2 bv = *(const float2*)&sB[laneLo * 68 + c0]; // ds_load_b64
      v2f bfrag; bfrag[0] = bv.x; bfrag[1] = bv.y;
      // two accumulator chains share each B fragment
      acc0 = __builtin_amdgcn_wmma_f32_16x16x4_f32(
          false, afragA[kk], false, bfrag, (short)0, acc0, false, false);
      acc1 = __builtin_amdgcn_wmma_f32_16x16x4_f32(
          false, afragB[kk], false, bfrag, (short)0, acc1, false, false);
    }

    // D layout: VGPR r -> M = r + 8*laneHi, N = laneLo  => j = j0 + laneLo
    // Within a lane j is monotonically increasing, so strict '<' keeps the
    // first-occurrence argmin (no equality tie-break needed here).
    const float wnv = wn[j0 + laneLo];
    const int   j   = j0 + laneLo;
#pragma unroll
    for (int r = 0; r < 8; ++r) {
      const float d0 = fmaf(-2.0f, acc0[r], wnv);
      if (d0 < minv0[r]) { minv0[r] = d0; mini0[r] = j; }
      const float d1 = fmaf(-2.0f, acc1[r], wnv);
      if (d1 < minv1[r]) { minv1[r] = d1; mini1[r] = j; }
    }
  }

  // --- Butterfly argmin across the 16 lanes of each half (wave32 shuffles).
  //     Ties across lanes are distinct j -> pick the smaller index. ---
#pragma unroll
  for (int r = 0; r < 8; ++r) {
#pragma unroll
    for (int mask = 1; mask <= 8; mask <<= 1) {
      float ov = __shfl_xor(minv0[r], mask, 32);
      int   oi = __shfl_xor(mini0[r], mask, 32);
      if (ov < minv0[r] || (ov == minv0[r] && oi < mini0[r])) {
        minv0[r] = ov; mini0[r] = oi;
      }
      ov = __shfl_xor(minv1[r], mask, 32);
      oi = __shfl_xor(mini1[r], mask, 32);
      if (ov < minv1[r] || (ov == minv1[r] && oi < mini1[r])) {
        minv1[r] = ov; mini1[r] = oi;
      }
    }
  }

  if (laneLo == 0) {
#pragma unroll
    for (int r = 0; r < 8; ++r) {
      const int m = r + 8 * laneHi;        // row within the M tile
      idxOut[nBase + m] = mini0[r];
      atomicAdd(&hist[mini0[r]], 1);
      idxOut[nBase + 16 + m] = mini1[r];
      atomicAdd(&hist[mini1[r]], 1);
    }
  }
}

// ---------------------------------------------------------------------------
// Kernel 3: codebook gather -> z_q (BCHW), accumulate sum((z_q - z)^2).
// 65536 threads, one per pixel-vector.
// ---------------------------------------------------------------------------
__global__ void __launch_bounds__(256)
vq_output(const float* __restrict__ z,
          const float* __restrict__ w,
          const int* __restrict__ idx,
          float* __restrict__ out,        // out[0]=loss, out[1..]=z_q
          float* __restrict__ lossAcc) {
  const int n = blockIdx.x * blockDim.x + threadIdx.x;
  const int bimg = n >> 10;
  const int hw   = n & (HW - 1);
  const int id   = idx[n];
  const float4* wr = (const float4*)(w + (size_t)id * C_DIM);
  const float* zp = z + ((size_t)bimg * C_DIM) * HW + hw;
  float* op = out + 1 + ((size_t)bimg * C_DIM) * HW + hw;

  float acc = 0.0f;
#pragma unroll
  for (int c4 = 0; c4 < C_DIM / 4; ++c4) {
    const float4 wq = wr[c4];              // contiguous: global_load_b128
    const float q[4] = {wq.x, wq.y, wq.z, wq.w};
#pragma unroll
    for (int k = 0; k < 4; ++k) {
      const int c = c4 * 4 + k;
      const float zv = zp[(size_t)c * HW]; // coalesced across threads
      op[(size_t)c * HW] = q[k];
      const float d = q[k] - zv;
      acc = fmaf(d, d, acc);
    }
  }

  __shared__ float red[256];
  red[threadIdx.x] = acc;
  __syncthreads();
#pragma unroll
  for (int s = 128; s > 0; s >>= 1) {
    if (threadIdx.x < s) red[threadIdx.x] += red[threadIdx.x + s];
    __syncthreads();
  }
  if (threadIdx.x == 0) atomicAdd(lossAcc, red[0]);
}

// ---------------------------------------------------------------------------
// Kernel 4: perplexity from histogram + final loss scale. One 1024-thread block.
// ---------------------------------------------------------------------------
__global__ void __launch_bounds__(1024)
vq_finalize(const int* __restrict__ hist,
            const float* __restrict__ lossAcc,
            float* __restrict__ out) {
  __shared__ float red[1024];
  const int t = threadIdx.x;
  const float e = (float)hist[t] * (1.0f / (float)N_VEC);
  red[t] = e * logf(e + 1e-10f);
  __syncthreads();
#pragma unroll
  for (int s = 512; s > 0; s >>= 1) {
    if (t < s) red[t] += red[t + s];
    __syncthreads();
  }
  if (t == 0) {
    out[0] = 1.25f * lossAcc[0] * (1.0f / (float)TOT_ELEM);  // (1+BETA)*MSE
    out[1 + TOT_ELEM] = expf(-red[0]);
  }
}

// ---------------------------------------------------------------------------
extern "C" void kernel_launch(void* const* d_in, const int* in_sizes, int n_in,
                              void* d_out, int out_size, void* d_ws, size_t ws_size,
                              hipStream_t stream) {
  const float* z = (const float*)d_in[0];   // 4194304 f32
  const float* w = (const float*)d_in[1];   // 65536 f32
  float* out = (float*)d_out;               // 4194306 f32

  // Workspace layout (bytes):
  char* ws = (char*)d_ws;
  float* wt      = (float*)(ws);                     // 64*1024 f32 = 256 KB
  float* wn      = (float*)(ws + 262144);            // 1024 f32
  int*   hist    = (int*)  (ws + 266240);            // 1024 i32
  int*   idx     = (int*)  (ws + 270336);            // 65536 i32 = 256 KB
  float* lossAcc = (float*)(ws + 532480);            // 1 f32

  vq_prep<<<4, 256, 0, stream>>>(w, wt, wn, hist, lossAcc);
  vq_argmin<<<256, 256, 0, stream>>>(z, wt, wn, idx, hist);
  vq_output<<<256, 256, 0, stream>>>(z, w, idx, out, lossAcc);
  vq_finalize<<<1, 1024, 0, stream>>>(hist, lossAcc, out);
}